// GraphConvolution_60266981097531
// MI455X (gfx1250) — compile-verified
//
#include <hip/hip_runtime.h>
#include <hip/hip_bf16.h>

// GCN layer for MI455X (gfx1250, wave32, WMMA).
// B=8 H=8 N=1024 D_IN=D_OUT=64.
//
//   deg[n]      = rsqrt(sum_m adj[n,m])
//   support2    = deg[m] * (X @ W[h])                (fp32 WMMA 16x16x4)
//   out[n,:]    = deg[n] * (adj @ support2) + bias   (fp32 WMMA 16x16x4)
//
// adj (256 MB) is read exactly twice (deg pass + GEMM pass). support2 is
// staged through 64 KB LDS per workgroup in 4 K-phases of 256 rows using the
// CDNA5 async global->LDS scatter path (GLOBAL_LOAD_ASYNC_TO_LDS_B32,
// ASYNCcnt), pre-swizzled into the wave32 B-fragment layout so the hot loop
// is 1x global_load_b64 + 4x ds_load_b64 + 4x v_wmma per K-step.

typedef __attribute__((ext_vector_type(2))) float v2f;
typedef __attribute__((ext_vector_type(8))) float v8f;

#define BH   64          // B*H
#define NN   1024
#define DD   64

#if __has_builtin(__builtin_amdgcn_global_load_async_to_lds_b32)
#define GCN_HAVE_ASYNC_LDS 1
// Builtin signature (from hipcc diagnostic): (AS1 int*, AS3 int*, imm, imm)
typedef __attribute__((address_space(1))) int* gip;
typedef __attribute__((address_space(3))) int* sip;
#endif

static __device__ __forceinline__ void wait_asynccnt0() {
#if __has_builtin(__builtin_amdgcn_s_wait_asynccnt)
    __builtin_amdgcn_s_wait_asynccnt(0);
#else
    asm volatile("s_wait_asynccnt 0x0" ::: "memory");
#endif
}

// ---------------------------------------------------------------- deg kernel
__global__ __launch_bounds__(256) void gcn_deg_kernel(
    const float* __restrict__ adj, float* __restrict__ deg)
{
    const int lane = threadIdx.x & 31;
    const int wave = threadIdx.x >> 5;                 // 8 waves / block
    const long row = (long)blockIdx.x * 8 + wave;      // 65536 rows total
    const float* p = adj + row * NN;
    float s = 0.f;
    #pragma unroll 8
    for (int j = lane; j < NN; j += 32) s += p[j];     // coalesced
    #pragma unroll
    for (int off = 16; off > 0; off >>= 1) s += __shfl_xor(s, off, 32);
    if (lane == 0) deg[row] = rsqrtf(s);
}

// ------------------------------------------------- support2 = deg[m]*(X@W[h])
__global__ __launch_bounds__(256) void gcn_support_kernel(
    const float* __restrict__ X, const float* __restrict__ W,
    const float* __restrict__ deg, float* __restrict__ sup)
{
    const int lane = threadIdx.x & 31;
    const int wave = threadIdx.x >> 5;                 // 8 waves / block
    const int bh   = blockIdx.x >> 3;                  // 8 blocks per bh
    const int h    = bh & 7;
    const int m0   = (blockIdx.x & 7) * 128 + wave * 16;
    const int lo   = lane & 15, hi = lane >> 4;

    // A fragment source: X rows m0+lo, cols 2*hi + 4*ks (+0,+1)
    const float* xrow = X + (size_t)bh * (NN * DD) + (size_t)(m0 + lo) * DD + 2 * hi;
    const float* wp   = W + h * (DD * DD);

    v8f acc[4] = {};
    for (int ks = 0; ks < 16; ++ks) {                  // K = 64
        v2f a = *(const v2f*)(xrow + 4 * ks);
        const int kb = 4 * ks + 2 * hi;
        #pragma unroll
        for (int t = 0; t < 4; ++t) {                  // 4 N-tiles -> N=64
            const int e = t * 16 + lo;
            v2f b;
            b.x = wp[kb * DD + e];
            b.y = wp[kb * DD + DD + e];
            acc[t] = __builtin_amdgcn_wmma_f32_16x16x4_f32(
                false, a, false, b, (short)0, acc[t], false, false);
        }
    }

    // scale rows by deg[m] and store
    float dm[8];
    #pragma unroll
    for (int v = 0; v < 8; ++v) dm[v] = deg[bh * NN + m0 + hi * 8 + v];
    float* op = sup + (size_t)bh * (NN * DD);
    #pragma unroll
    for (int t = 0; t < 4; ++t)
        #pragma unroll
        for (int v = 0; v < 8; ++v)
            op[(size_t)(m0 + hi * 8 + v) * DD + t * 16 + lo] = acc[t][v] * dm[v];
}

// ---------------------------------------- out = deg[n]*(adj @ support2)+bias
__global__ __launch_bounds__(512) void gcn_main_kernel(
    const float* __restrict__ adj, const float* __restrict__ sup,
    const float* __restrict__ deg, const float* __restrict__ bias,
    float* __restrict__ out)
{
    // 64 K-steps * 4 e-tiles * 32 lanes * 2 floats = 16384 floats = 64 KB
    __shared__ float lds[16384];

    const int lane  = threadIdx.x & 31;
    const int wave  = threadIdx.x >> 5;                // 16 waves
    const int bh    = blockIdx.x >> 2;                 // 4 blocks per bh
    const int chunk = blockIdx.x & 3;                  // 256 rows each
    const int n0    = chunk * 256 + wave * 16;
    const int lo    = lane & 15, hi = lane >> 4;

    const float* arow = adj + (size_t)bh * (NN * NN) + (size_t)(n0 + lo) * NN + 2 * hi;
    const float* sp   = sup + (size_t)bh * (NN * DD);

    v8f acc[4] = {};

    for (int p = 0; p < 4; ++p) {                      // 4 K-phases of 256
        __syncthreads();                               // lds reuse safety
        // Stage support2 rows [p*256, p*256+256) into LDS, swizzled so a
        // B-fragment (4x16 f32) is one contiguous float2 per lane:
        //   element (k,e): ks=k>>2, kr=k&3, lane=((kr>>1)<<4)|(e&15), slot=kr&1
        // Global side is fully coalesced (consecutive threads -> consecutive e);
        // the LDS-side scatter is done by the async copy's per-lane LDS address.
        #pragma unroll 4
        for (int i = 0; i < 32; ++i) {                 // 16384 / 512 threads
            const int idx = threadIdx.x + 512 * i;
            const int kl  = idx >> 6;                  // 0..255
            const int e   = idx & 63;
            const int ks = kl >> 2, kr = kl & 3;
            const int dl = ((kr >> 1) << 4) | (e & 15);
            const int dst = ((((ks << 2) + (e >> 4)) * 32 + dl) << 1) | (kr & 1);
            const float* g = sp + (size_t)(p * 256 + kl) * DD + e;
#ifdef GCN_HAVE_ASYNC_LDS
            __builtin_amdgcn_global_load_async_to_lds_b32(
                (gip)(float*)g, (sip)(lds + dst), 0, 0);   // ASYNCcnt path
#else
            lds[dst] = *g;                             // sync fallback
#endif
        }
#ifdef GCN_HAVE_ASYNC_LDS
        wait_asynccnt0();                              // this wave's copies done
#endif
        __syncthreads();                               // all waves' copies visible

        const v2f* bfrag = (const v2f*)lds;
        const float* ap  = arow + p * 256;
        #pragma unroll 4
        for (int ks = 0; ks < 64; ++ks) {
            v2f a = *(const v2f*)(ap + 4 * ks);        // global_load_b64 (adj)
            #pragma unroll
            for (int t = 0; t < 4; ++t) {
                v2f b = bfrag[((ks << 2) + t) * 32 + lane];  // ds_load_b64
                acc[t] = __builtin_amdgcn_wmma_f32_16x16x4_f32(
                    false, a, false, b, (short)0, acc[t], false, false);
            }
        }
    }

    // epilogue: out = deg[n]*acc + bias
    float dn[8];
    #pragma unroll
    for (int v = 0; v < 8; ++v) dn[v] = deg[bh * NN + n0 + hi * 8 + v];
    float* op = out + (size_t)bh * (NN * DD);
    #pragma unroll
    for (int t = 0; t < 4; ++t) {
        const float bv = bias[t * 16 + lo];
        #pragma unroll
        for (int v = 0; v < 8; ++v)
            op[(size_t)(n0 + hi * 8 + v) * DD + t * 16 + lo] = acc[t][v] * dn[v] + bv;
    }
}

// ------------------------------------------------------------------- launch
extern "C" void kernel_launch(void* const* d_in, const int* in_sizes, int n_in,
                              void* d_out, int out_size, void* d_ws, size_t ws_size,
                              hipStream_t stream)
{
    const float* input  = (const float*)d_in[0];   // [8,8,1024,64]
    const float* adj    = (const float*)d_in[1];   // [8,8,1024,1024]
    const float* weight = (const float*)d_in[2];   // [8,64,64]
    const float* bias   = (const float*)d_in[3];   // [64]
    float* out = (float*)d_out;                    // [8,8,1024,64]

    float* deg = (float*)d_ws;                     // 64*1024 floats
    float* sup = deg + BH * NN;                    // 64*1024*64 floats (~16 MB)

    gcn_deg_kernel    <<<BH * NN / 8, 256, 0, stream>>>(adj, deg);
    gcn_support_kernel<<<BH * 8,      256, 0, stream>>>(input, weight, deg, sup);
    gcn_main_kernel   <<<BH * 4,      512, 0, stream>>>(adj, sup, deg, bias, out);
}